// DirectionalContrastiveLoss_49684181680813
// MI455X (gfx1250) — compile-verified
//
#include <hip/hip_runtime.h>
#include <math.h>

// Problem constants (from reference): N = M = 8000 anchors/memory, C = 256.
#define N_ANCH 8000
#define CDIM   256
#define MMEM   8000
#define SEL    4000
#define HW     4000           // h*w
#define INV_TEMP 10.0f        // 1/0.1
#define CEPS   1e-8f

typedef float v2f __attribute__((ext_vector_type(2)));
typedef float v8f __attribute__((ext_vector_type(8)));

#define ASTRIDE 260           // padded LDS row stride (floats); 260 % 64 = 4 -> conflict-free
#define MSTRIDE 260
#define ROWS_PER_WG 32        // 2 waves x 16 anchor rows
#define TILE_N 16             // memory columns per tile

__global__ void dcl_init_accum(float* accum) {
    if (threadIdx.x < 4) accum[threadIdx.x] = 0.0f;
}

// Build memory matrix [M][C] and memory_labels [M].
// ul layout is (b, c, h, w); flat anchor n = bi*HW + rem maps to ul[bi*C*HW + ch*HW + rem].
__global__ void dcl_gather(const float* __restrict__ ul1, const float* __restrict__ ul2,
                           const int* __restrict__ idx1, const int* __restrict__ idx2,
                           const int* __restrict__ pl1, const int* __restrict__ pl2,
                           float* __restrict__ mem, int* __restrict__ memLab) {
    int t = blockIdx.x * blockDim.x + threadIdx.x;
    if (t >= MMEM * CDIM) return;
    int j  = t >> 8;          // memory row
    int ch = t & 255;         // channel
    int src;
    const float* ul;
    if (j < SEL) { src = idx1[j];       ul = ul1; }
    else         { src = idx2[j - SEL]; ul = ul2; }
    int bi  = src / HW;
    int rem = src - bi * HW;
    mem[t] = ul[bi * CDIM * HW + ch * HW + rem];
    if (ch == 0) memLab[j] = (j < SEL) ? pl1[src] : pl2[src];
}

// pos[i] = dot(f1_i, f2_i)/TEMP (identical value for both directions), masks + mask counts.
__global__ void dcl_pos_mask(const float* __restrict__ f1, const float* __restrict__ f2,
                             const float* __restrict__ pg1, const float* __restrict__ pg2,
                             float* __restrict__ pos, float* __restrict__ mask1,
                             float* __restrict__ mask2, float* __restrict__ accum) {
    int i = blockIdx.x * blockDim.x + threadIdx.x;
    if (i >= N_ANCH) return;
    const float4* a = (const float4*)(f1 + (size_t)i * CDIM);
    const float4* b = (const float4*)(f2 + (size_t)i * CDIM);
    float d = 0.0f;
#pragma unroll 8
    for (int k = 0; k < CDIM / 4; ++k) {
        float4 x = a[k], y = b[k];
        d += x.x * y.x + x.y * y.y + x.z * y.z + x.w * y.w;
    }
    pos[i] = d * INV_TEMP;
    float g1 = pg1[i], g2 = pg2[i];
    float m1 = (g2 > 0.7f && g1 < g2) ? 1.0f : 0.0f;
    float m2 = (g1 > 0.7f && g2 < g1) ? 1.0f : 0.0f;
    mask1[i] = m1;
    mask2[i] = m2;
    if (m1 != 0.0f) atomicAdd(&accum[2], 1.0f);
    if (m2 != 0.0f) atomicAdd(&accum[3], 1.0f);
}

// Fused GEMM (v_wmma_f32_16x16x4_f32) + masked online softmax + loss accumulation.
// One workgroup = 2 waves; each wave owns 16 anchor rows (tile of 32 per WG staged in LDS).
// Streams memory in 16-column tiles; per-lane online (max,sum), final 16-lane shuffle reduce.
__global__ __launch_bounds__(64) void dcl_contrastive(
    const float* __restrict__ anchor, const int* __restrict__ ancLab,
    const float* __restrict__ pos, const float* __restrict__ pmask,
    const float* __restrict__ mem, const int* __restrict__ memLab,
    float* __restrict__ lossAccum) {
    extern __shared__ float lds[];
    float* aMat  = lds;                              // 32 x ASTRIDE
    float* mTile = lds + ROWS_PER_WG * ASTRIDE;      // 16 x MSTRIDE

    const int tid  = threadIdx.x;
    const int wave = tid >> 5;
    const int lane = tid & 31;
    const int half = lane >> 4;       // 0: k-offsets {0,1}, rows r; 1: k-offsets {2,3}, rows r+8
    const int l16  = lane & 15;
    const int base = blockIdx.x * ROWS_PER_WG;

    // Stage the 32x256 anchor block into LDS (padded stride).
    for (int t = tid; t < ROWS_PER_WG * (CDIM / 4); t += 64) {
        int row = t >> 6;             // /64 float4 per row
        int c4  = t & 63;
        float4 v = *(const float4*)(anchor + (size_t)(base + row) * CDIM + c4 * 4);
        *(float4*)(aMat + row * ASTRIDE + c4 * 4) = v;
    }

    // Reference's transposed mask: row i uses memory_labels[i], column j uses anchor_labels[j].
    int rml[8];
#pragma unroll
    for (int r = 0; r < 8; ++r)
        rml[r] = memLab[base + wave * 16 + half * 8 + r];

    float mrun[8], srun[8];
#pragma unroll
    for (int r = 0; r < 8; ++r) { mrun[r] = -INFINITY; srun[r] = 0.0f; }

    const float* aBase = aMat + (wave * 16 + l16) * ASTRIDE + half * 2;

    for (int ct = 0; ct < MMEM / TILE_N; ++ct) {
        __syncthreads();
        for (int t = tid; t < TILE_N * (CDIM / 4); t += 64) {
            int row = t >> 6;
            int c4  = t & 63;
            float4 v = *(const float4*)(mem + (size_t)(ct * TILE_N + row) * CDIM + c4 * 4);
            *(float4*)(mTile + row * MSTRIDE + c4 * 4) = v;
        }
        if (ct + 1 < MMEM / TILE_N)   // warm L2->L0 for the next memory tile
            __builtin_prefetch(mem + (size_t)(ct + 1) * TILE_N * CDIM + tid * 16, 0, 0);
        int colLab = ancLab[ct * TILE_N + l16];
        __syncthreads();

        const float* bBase = mTile + l16 * MSTRIDE + half * 2;
        v8f acc = {0.f, 0.f, 0.f, 0.f, 0.f, 0.f, 0.f, 0.f};
#pragma unroll 8
        for (int k0 = 0; k0 < CDIM; k0 += 4) {
            v2f a = *(const v2f*)(aBase + k0);
            v2f b = *(const v2f*)(bBase + k0);
            acc = __builtin_amdgcn_wmma_f32_16x16x4_f32(false, a, false, b,
                                                        (short)0, acc, false, false);
        }

        // Per-lane online softmax update (max is over ALL columns, sum is masked).
#pragma unroll
        for (int r = 0; r < 8; ++r) {
            float v  = acc[r] * INV_TEMP;
            float mk = (rml[r] != colLab) ? 1.0f : 0.0f;
            float nm = fmaxf(mrun[r], v);
            srun[r]  = srun[r] * __expf(mrun[r] - nm) + mk * __expf(v - nm);
            mrun[r]  = nm;
        }
    }

    // Row reductions across the 16 lanes of each half-wave, then fold in the pos column.
    float lsum = 0.0f;
#pragma unroll
    for (int r = 0; r < 8; ++r) {
        float m = mrun[r];
        float rowmax = m;
#pragma unroll
        for (int off = 1; off < 16; off <<= 1)
            rowmax = fmaxf(rowmax, __shfl_xor(rowmax, off, 16));
        float p = srun[r] * __expf(m - rowmax);
#pragma unroll
        for (int off = 1; off < 16; off <<= 1)
            p += __shfl_xor(p, off, 16);
        if (l16 == 0) {
            int i      = base + wave * 16 + half * 8 + r;
            float p0   = pos[i];
            float fm   = fmaxf(rowmax, p0);                       // pos joins the row max
            float tot  = p * __expf(rowmax - fm) + __expf(p0 - fm); // pos column mask == 1
            float lgt  = __expf(p0 - fm) / (tot + CEPS);
            lsum      += -__logf(lgt + CEPS) * pmask[i];
        }
    }
    if (l16 == 0) atomicAdd(lossAccum, lsum);
}

__global__ void dcl_finalize(const float* __restrict__ accum, float* __restrict__ out) {
    out[0] = accum[0] / (accum[2] + 1e-12f) + accum[1] / (accum[3] + 1e-12f);
}

extern "C" void kernel_launch(void* const* d_in, const int* in_sizes, int n_in,
                              void* d_out, int out_size, void* d_ws, size_t ws_size,
                              hipStream_t stream) {
    const float* f1  = (const float*)d_in[0];
    const float* f2  = (const float*)d_in[1];
    const int*   pl1 = (const int*)d_in[2];
    const int*   pl2 = (const int*)d_in[3];
    const float* pg1 = (const float*)d_in[4];
    const float* pg2 = (const float*)d_in[5];
    const float* ul1 = (const float*)d_in[6];
    const float* ul2 = (const float*)d_in[7];
    const int*   idx1 = (const int*)d_in[8];
    const int*   idx2 = (const int*)d_in[9];

    // Workspace layout (~8.4 MB): memory matrix, pos, masks, accumulators, labels.
    float* ws     = (float*)d_ws;
    float* mem    = ws;                              // M*C
    float* pos    = mem + (size_t)MMEM * CDIM;       // N
    float* mask1  = pos + N_ANCH;                    // N
    float* mask2  = mask1 + N_ANCH;                  // N
    float* accum  = mask2 + N_ANCH;                  // [num1, num2, cnt1, cnt2]
    int*   memLab = (int*)(accum + 8);               // M

    dcl_init_accum<<<1, 32, 0, stream>>>(accum);
    dcl_gather<<<(MMEM * CDIM) / 256, 256, 0, stream>>>(ul1, ul2, idx1, idx2,
                                                        pl1, pl2, mem, memLab);
    dcl_pos_mask<<<(N_ANCH + 255) / 256, 256, 0, stream>>>(f1, f2, pg1, pg2,
                                                           pos, mask1, mask2, accum);

    size_t smem = (size_t)(ROWS_PER_WG * ASTRIDE + TILE_N * MSTRIDE) * sizeof(float);
    // Direction 1: anchors = f1, column labels = pseudo_label1
    dcl_contrastive<<<N_ANCH / ROWS_PER_WG, 64, smem, stream>>>(
        f1, pl1, pos, mask1, mem, memLab, &accum[0]);
    // Direction 2: anchors = f2, column labels = pseudo_label2
    dcl_contrastive<<<N_ANCH / ROWS_PER_WG, 64, smem, stream>>>(
        f2, pl2, pos, mask2, mem, memLab, &accum[1]);

    dcl_finalize<<<1, 1, 0, stream>>>(accum, (float*)d_out);
}